// MoeMatmul_39453569581158
// MI455X (gfx1250) — compile-verified
//
#include <hip/hip_runtime.h>
#include <stdint.h>

// MoE expert GEMM: out[T, D_OUT] = state[T, D_IN] * w[expert]^T, fp32.
// Bandwidth-bound (~384 MB traffic @ 23.3 TB/s ~= 16.5 us vs <1 us of WMMA math).
// Strategy: stream weights once via GLOBAL_LOAD_ASYNC_TO_LDS_B128 (double
// buffered, ASYNCcnt-tracked), keep state in L2, V_WMMA_F32_16X16X4_F32 math.

#define D_IN   4096
#define D_OUT  14336
#define TOKENS 2048

#define M_BLK  128                 // tokens per block
#define N_BLK  128                 // d_out per block
#define K_BLK  32                  // k per stage
#define NCHUNK (D_IN / K_BLK)      // 128 pipeline stages
#define LDS_STRIDE (K_BLK + 4)     // 36 floats = 144 B rows: 16B aligned, conflict-free

typedef __attribute__((ext_vector_type(2))) float v2f;
typedef __attribute__((ext_vector_type(8))) float v8f;

// Async DMA: global -> LDS, 16 B per lane, tracked by ASYNCcnt.
// GVS addressing: mem = SGPR64 base + per-lane i32 byte offset.
__device__ __forceinline__ void async_b128(uint32_t lds_addr, int byte_off,
                                           const float* __restrict__ sbase) {
    asm volatile("global_load_async_to_lds_b128 %0, %1, %2"
                 :
                 : "v"(lds_addr), "v"(byte_off), "s"(sbase)
                 : "memory");
}

__device__ __forceinline__ void wait_async(int pending) {
    if (pending) asm volatile("s_wait_asynccnt 0x8" ::: "memory");
    else         asm volatile("s_wait_asynccnt 0x0" ::: "memory");
}

__global__ __launch_bounds__(256)
void moe_gemm_wmma_f32(const float* __restrict__ state,
                       const int*   __restrict__ expert_id,
                       const float* __restrict__ w,
                       float*       __restrict__ out)
{
    __shared__ __align__(16) float As[2][M_BLK * LDS_STRIDE];   // 2 x 18 KB
    __shared__ __align__(16) float Bs[2][N_BLK * LDS_STRIDE];   // 2 x 18 KB

    const int tid  = threadIdx.x;
    const int lane = tid & 31;
    const int wave = tid >> 5;          // 0..7
    const int half = lane >> 4;         // lanes 0-15 -> K={0,1}; 16-31 -> K={2,3}
    const int r    = lane & 15;

    const int n0 = blockIdx.x * N_BLK;  // d_out tile origin
    const int m0 = blockIdx.y * M_BLK;  // token tile origin

    const int wm = wave & 1;            // 2 wave-rows * 64 tokens
    const int wn = wave >> 1;           // 4 wave-cols * 32 outs

    const long long e = (long long)expert_id[0];
    const float* __restrict__ wexp = w + e * (long long)D_OUT * (long long)D_IN;

    // Per-thread staging coords: 8 consecutive threads cover one 128 B row chunk.
    // idx = tid + it*256 ; row = idx/8 ; c4 = (idx%8)*4 floats.
    const uint32_t a_base0 = (uint32_t)(uintptr_t)&As[0][0];
    const uint32_t a_base1 = (uint32_t)(uintptr_t)&As[1][0];
    const uint32_t b_base0 = (uint32_t)(uintptr_t)&Bs[0][0];
    const uint32_t b_base1 = (uint32_t)(uintptr_t)&Bs[1][0];

    auto issue_chunk = [&](int buf, int k0) {
        const uint32_t abase = buf ? a_base1 : a_base0;
        const uint32_t bbase = buf ? b_base1 : b_base0;
        #pragma unroll
        for (int it = 0; it < 4; ++it) {
            const int idx = tid + it * 256;
            const int row = idx >> 3;
            const int c4  = (idx & 7) << 2;
            const uint32_t lds_off = (uint32_t)((row * LDS_STRIDE + c4) << 2);
            // Byte offsets fit i32: expert slice = 235 MB, state = 32 MB.
            const int aoff = ((m0 + row) * D_IN + k0 + c4) << 2;
            const int boff = ((n0 + row) * D_IN + k0 + c4) << 2;
            async_b128(abase + lds_off, aoff, state);   // ASYNCcnt += 1 (wave)
            async_b128(bbase + lds_off, boff, wexp);    // ASYNCcnt += 1 (wave)
        }
    };

    v8f acc[4][2];
    #pragma unroll
    for (int mi = 0; mi < 4; ++mi)
        #pragma unroll
        for (int ni = 0; ni < 2; ++ni)
            acc[mi][ni] = (v8f)0.0f;

    // Prologue: fill buffer 0.
    issue_chunk(0, 0);

    for (int chunk = 0; chunk < NCHUNK; ++chunk) {
        const int cur = chunk & 1;
        const bool more = (chunk + 1) < NCHUNK;

        // Kick off next chunk's DMA into the other buffer (safe: trailing
        // barrier of the previous iteration guarantees nobody reads it).
        if (more) issue_chunk(cur ^ 1, (chunk + 1) * K_BLK);

        // Wait for *this* chunk's 8 async ops (issued last iteration) to land;
        // the 8 just issued may remain in flight.
        wait_async(more ? 8 : 0);
        __syncthreads();   // make LDS writes visible to all waves

        const float* __restrict__ Ab = &As[cur][0];
        const float* __restrict__ Bb = &Bs[cur][0];

        // ---- 8 WMMA k-steps over this K chunk ----
        #pragma unroll
        for (int k4 = 0; k4 < K_BLK; k4 += 4) {
            const int kc = k4 + 2 * half;   // even -> 8B-aligned float2

            v2f afrag[4];
            #pragma unroll
            for (int mi = 0; mi < 4; ++mi)
                afrag[mi] = *(const v2f*)(Ab + (wm * 64 + mi * 16 + r) * LDS_STRIDE + kc);

            v2f bfrag[2];
            #pragma unroll
            for (int ni = 0; ni < 2; ++ni)
                bfrag[ni] = *(const v2f*)(Bb + (wn * 32 + ni * 16 + r) * LDS_STRIDE + kc);

            #pragma unroll
            for (int mi = 0; mi < 4; ++mi)
                #pragma unroll
                for (int ni = 0; ni < 2; ++ni)
                    acc[mi][ni] = __builtin_amdgcn_wmma_f32_16x16x4_f32(
                        false, afrag[mi], false, bfrag[ni],
                        (short)0, acc[mi][ni], false, false);
        }
        __syncthreads();   // all waves done with buffer `cur` before it is re-filled
    }

    // ---- epilogue: C/D layout is VGPR i -> M = i + 8*half, N = r ----
    #pragma unroll
    for (int mi = 0; mi < 4; ++mi) {
        #pragma unroll
        for (int ni = 0; ni < 2; ++ni) {
            const int col = n0 + wn * 32 + ni * 16 + r;
            #pragma unroll
            for (int i = 0; i < 8; ++i) {
                const int row = m0 + wm * 64 + mi * 16 + i + 8 * half;
                out[(long long)row * D_OUT + col] = acc[mi][ni][i];
            }
        }
    }
}

extern "C" void kernel_launch(void* const* d_in, const int* in_sizes, int n_in,
                              void* d_out, int out_size, void* d_ws, size_t ws_size,
                              hipStream_t stream) {
    const float* state     = (const float*)d_in[0];
    const int*   expert_id = (const int*)  d_in[1];
    const float* w         = (const float*)d_in[2];
    float*       out       = (float*)d_out;

    dim3 grid(D_OUT / N_BLK, TOKENS / M_BLK);   // 112 x 16 blocks
    moe_gemm_wmma_f32<<<grid, dim3(256), 0, stream>>>(state, expert_id, w, out);
}